// GCN_15693810499984
// MI455X (gfx1250) — compile-verified
//
#include <hip/hip_runtime.h>

typedef float v2f __attribute__((ext_vector_type(2)));
typedef float v8f __attribute__((ext_vector_type(8)));
typedef unsigned int u32x4 __attribute__((ext_vector_type(4)));
typedef int i32x4 __attribute__((ext_vector_type(4)));
typedef int i32x8 __attribute__((ext_vector_type(8)));

#define N_NODES 100000
#define N_EDGES 3200000
#define N_FEAT  512
#define N_HID   128
#define N_CLASS 40
#define M_TILES (N_NODES / 16)   // 6250, exact

// ---------------------------------------------------------------------------
// TDM: contiguous copy of `nelems` f32 from global to LDS as a 1-D tile.
// D# layout per CDNA5 ISA ch.8 (group0: count/lds_addr/global_addr/type=2,
// group1: data_size=4B, tensor_dim0=nelems, tensor_dim1=1, tile_dim0=nelems,
// tile_dim1=1, dim0_stride=nelems; groups 2/3 zero -> dims 2..4 unused).
// Toolchain uses the 6-arg builtin: (u32x4, i32x8, i32x4, i32x4, i32x8, cpol).
// ---------------------------------------------------------------------------
static __device__ __forceinline__ void tdm_copy_f32_to_lds(
    void* lds_dst, const float* src, unsigned nelems)
{
    const unsigned long long ga = (unsigned long long)(uintptr_t)src;
    const unsigned lds_off = (unsigned)(uintptr_t)lds_dst; // low 32b = LDS byte offset

    u32x4 g0;
    g0.x = 0x1u;                                            // count=1 (valid), user D#
    g0.y = lds_off;                                         // lds_addr
    g0.z = (unsigned)ga;                                    // global_addr[31:0]
    g0.w = (unsigned)((ga >> 32) & 0x01FFFFFFu) | (2u << 30); // addr[56:32] | type=2

    i32x8 g1;
    g1[0] = (2 << 16);                                      // data_size = 4 bytes
    g1[1] = (int)((nelems & 0xFFFFu) << 16);                // tensor_dim0[15:0]
    g1[2] = (int)(((nelems >> 16) & 0xFFFFu) | (1u << 16)); // tensor_dim0[31:16] | tensor_dim1=1
    g1[3] = (int)((nelems & 0xFFFFu) << 16);                // tile_dim0 = nelems
    g1[4] = 1;                                              // tile_dim1 = 1, tile_dim2 = 0
    g1[5] = (int)nelems;                                    // tensor_dim0_stride[31:0]
    g1[6] = 0;                                              // stride0[47:32] | stride1[15:0]
    g1[7] = 0;

    i32x4 gz4 = {0, 0, 0, 0};
    i32x8 gz8 = {0, 0, 0, 0, 0, 0, 0, 0};
    __builtin_amdgcn_tensor_load_to_lds(g0, g1, gz4, gz4, gz8, 0);
}

// ---------------------------------------------------------------------------
// GEMM1: H1[100000,128] = X[100000,512] @ W1[512,128] + b1
// 8 waves/block; wave w owns M-tile (blockIdx*8+w): 16 rows x 128 cols.
// W1 staged through LDS in 128-row K-chunks (64 KB), double-buffered via the
// Tensor Data Mover (wave 0 issues tensor_load_to_lds; s_wait_tensorcnt syncs).
// ---------------------------------------------------------------------------
__global__ __launch_bounds__(256) void gemm1_wmma(
    const float* __restrict__ X, const float* __restrict__ W1,
    const float* __restrict__ b1, float* __restrict__ H1)
{
    __shared__ float ldsW[2][128 * N_HID];        // 2 x 64 KB

    const int tid  = threadIdx.x;
    const int wave = tid >> 5;
    const int lane = tid & 31;
    const int half = lane >> 4;
    const int l15  = lane & 15;
    const int mt   = blockIdx.x * 8 + wave;
    const bool valid = (mt < M_TILES);
    const int row  = mt * 16 + l15;

    constexpr unsigned CHUNK = 128 * N_HID;       // 16384 floats per K-chunk

    if (wave == 0)                                // prefetch chunk 0
        tdm_copy_f32_to_lds(ldsW[0], W1, CHUNK);

    v8f acc[8] = {};

    for (int c = 0; c < 4; ++c) {
        const int cur = c & 1;
        __syncthreads();   // prior iteration's reads of the other buffer done
        if (wave == 0) {
            if (c + 1 < 4) {
                // prefetch next chunk, then wait until only it is in flight:
                // TDM ops are in-order per wave, so chunk c has landed.
                tdm_copy_f32_to_lds(ldsW[cur ^ 1],
                                    W1 + (size_t)(c + 1) * CHUNK, CHUNK);
                __builtin_amdgcn_s_wait_tensorcnt(1);
            } else {
                __builtin_amdgcn_s_wait_tensorcnt(0);
            }
        }
        __syncthreads();   // chunk c visible to all waves

        if (valid) {
            const float* arow = X + (size_t)row * N_FEAT + c * 128 + 2 * half;
            const float* wbuf = ldsW[cur];
            for (int ks = 0; ks < 128; ks += 4) {
                v2f a;
                a.x = arow[ks];           // X[row][kc+ks+2*half]
                a.y = arow[ks + 1];
                const float* bbase = wbuf + (ks + 2 * half) * N_HID + l15;
                #pragma unroll
                for (int nt = 0; nt < 8; ++nt) {
                    v2f b;
                    b.x = bbase[nt * 16];           // W1[kk  ][n0+l15]
                    b.y = bbase[nt * 16 + N_HID];   // W1[kk+1][n0+l15]
                    acc[nt] = __builtin_amdgcn_wmma_f32_16x16x4_f32(
                        false, a, false, b, (short)0, acc[nt], false, false);
                }
            }
        }
    }

    if (valid) {
        #pragma unroll
        for (int nt = 0; nt < 8; ++nt) {
            const int col = nt * 16 + l15;
            const float bias = b1[col];
            #pragma unroll
            for (int v = 0; v < 8; ++v) {
                const int r = mt * 16 + v + 8 * half;
                H1[(size_t)r * N_HID + col] = acc[nt][v] + bias;
            }
        }
    }
}

// ---------------------------------------------------------------------------
// GEMM2: O2[100000,40] = relu(S1)[100000,128] @ W2[128,40] + b2
// ReLU fused into the A-fragment load. W2 padded to 128x48 in LDS.
// ---------------------------------------------------------------------------
__global__ __launch_bounds__(256) void gemm2_wmma(
    const float* __restrict__ S1, const float* __restrict__ W2,
    const float* __restrict__ b2, float* __restrict__ O2)
{
    __shared__ float ldsW[N_HID * 48];            // 24 KB

    const int tid  = threadIdx.x;
    const int wave = tid >> 5;
    const int lane = tid & 31;
    const int half = lane >> 4;
    const int l15  = lane & 15;

    for (int i = tid; i < N_HID * 48; i += 256) ldsW[i] = 0.0f;
    __syncthreads();
    for (int i = tid; i < N_HID * N_CLASS; i += 256) {
        const int k = i / N_CLASS, c = i % N_CLASS;
        ldsW[k * 48 + c] = W2[i];
    }
    __syncthreads();

    const int mt = blockIdx.x * 8 + wave;
    if (mt >= M_TILES) return;
    const int row = mt * 16 + l15;

    v8f acc[3] = {};
    const float* arow = S1 + (size_t)row * N_HID + 2 * half;
    for (int ks = 0; ks < N_HID; ks += 4) {
        v2f a;
        a.x = fmaxf(arow[ks],     0.0f);   // fused ReLU
        a.y = fmaxf(arow[ks + 1], 0.0f);
        const float* bbase = ldsW + (ks + 2 * half) * 48 + l15;
        #pragma unroll
        for (int nt = 0; nt < 3; ++nt) {
            v2f b;
            b.x = bbase[nt * 16];
            b.y = bbase[nt * 16 + 48];
            acc[nt] = __builtin_amdgcn_wmma_f32_16x16x4_f32(
                false, a, false, b, (short)0, acc[nt], false, false);
        }
    }

    #pragma unroll
    for (int nt = 0; nt < 3; ++nt) {
        const int col = nt * 16 + l15;
        if (col < N_CLASS) {
            const float bias = b2[col];
            #pragma unroll
            for (int v = 0; v < 8; ++v) {
                const int r = mt * 16 + v + 8 * half;
                O2[(size_t)r * N_CLASS + col] = acc[nt][v] + bias;
            }
        }
    }
}

// ---------------------------------------------------------------------------
// SPMM over 128 feats: out[r,:] += val * H[c,:]   (one wave per edge)
// Gathers are L2-resident (H1 = 51.2 MB < 192 MB L2); scatters via f32 atomics.
// ---------------------------------------------------------------------------
__global__ __launch_bounds__(256) void spmm128_atomic(
    const int* __restrict__ rows, const int* __restrict__ cols,
    const float* __restrict__ vals, const float* __restrict__ H,
    float* __restrict__ out)
{
    const int wid = blockIdx.x * 8 + (threadIdx.x >> 5);
    if (wid >= N_EDGES) return;
    const int lane = threadIdx.x & 31;
    const int r = rows[wid];
    const int c = cols[wid];
    const float v = vals[wid];

    const float4 h4 = ((const float4*)(H + (size_t)c * N_HID))[lane];
    float* op = out + (size_t)r * N_HID + 4 * lane;
    unsafeAtomicAdd(op + 0, v * h4.x);
    unsafeAtomicAdd(op + 1, v * h4.y);
    unsafeAtomicAdd(op + 2, v * h4.z);
    unsafeAtomicAdd(op + 3, v * h4.w);
}

// SPMM over 40 feats (one wave per edge; lanes 0..31 + lanes 0..7 for 32..39)
__global__ __launch_bounds__(256) void spmm40_atomic(
    const int* __restrict__ rows, const int* __restrict__ cols,
    const float* __restrict__ vals, const float* __restrict__ H,
    float* __restrict__ out)
{
    const int wid = blockIdx.x * 8 + (threadIdx.x >> 5);
    if (wid >= N_EDGES) return;
    const int lane = threadIdx.x & 31;
    const int r = rows[wid];
    const int c = cols[wid];
    const float v = vals[wid];

    const float* hp = H + (size_t)c * N_CLASS;
    float* op = out + (size_t)r * N_CLASS;
    unsafeAtomicAdd(op + lane, v * hp[lane]);
    if (lane < 8)
        unsafeAtomicAdd(op + 32 + lane, v * hp[32 + lane]);
}

// ---------------------------------------------------------------------------
// Row-wise log_softmax over 40 classes: one wave per row, shuffle reductions.
// ---------------------------------------------------------------------------
__global__ __launch_bounds__(256) void logsoftmax40(
    const float* __restrict__ S, float* __restrict__ out)
{
    const int r = blockIdx.x * 8 + (threadIdx.x >> 5);
    if (r >= N_NODES) return;
    const int lane = threadIdx.x & 31;
    const float* p = S + (size_t)r * N_CLASS;

    const float x1 = p[lane];
    const float x2 = (lane < 8) ? p[32 + lane] : -__builtin_inff();

    float m = fmaxf(x1, x2);
    #pragma unroll
    for (int off = 16; off > 0; off >>= 1)
        m = fmaxf(m, __shfl_xor(m, off, 32));

    float s = expf(x1 - m) + ((lane < 8) ? expf(x2 - m) : 0.0f);
    #pragma unroll
    for (int off = 16; off > 0; off >>= 1)
        s += __shfl_xor(s, off, 32);

    const float ls = logf(s);
    out[(size_t)r * N_CLASS + lane] = x1 - m - ls;
    if (lane < 8)
        out[(size_t)r * N_CLASS + 32 + lane] = x2 - m - ls;
}

__global__ void zero_kernel(float* __restrict__ p, size_t n)
{
    size_t i = (size_t)blockIdx.x * blockDim.x + threadIdx.x;
    const size_t stride = (size_t)gridDim.x * blockDim.x;
    for (; i < n; i += stride) p[i] = 0.0f;
}

// ---------------------------------------------------------------------------
extern "C" void kernel_launch(void* const* d_in, const int* in_sizes, int n_in,
                              void* d_out, int out_size, void* d_ws, size_t ws_size,
                              hipStream_t stream)
{
    const float* x    = (const float*)d_in[0];
    const int*   erow = (const int*)  d_in[1];
    const int*   ecol = (const int*)  d_in[2];
    const float* eval = (const float*)d_in[3];
    const float* W1   = (const float*)d_in[4];
    const float* b1   = (const float*)d_in[5];
    const float* W2   = (const float*)d_in[6];
    const float* b2   = (const float*)d_in[7];
    float* out = (float*)d_out;

    float* ws = (float*)d_ws;
    float* H1 = ws;                               // 100000*128
    float* S1 = H1 + (size_t)N_NODES * N_HID;     // 100000*128
    float* O2 = S1 + (size_t)N_NODES * N_HID;     // 100000*40
    float* S2 = O2 + (size_t)N_NODES * N_CLASS;   // 100000*40

    zero_kernel<<<2048, 256, 0, stream>>>(S1, (size_t)N_NODES * N_HID);
    zero_kernel<<<1024, 256, 0, stream>>>(S2, (size_t)N_NODES * N_CLASS);

    const int mblocks = (M_TILES + 7) / 8;        // 782
    const int eblocks = (N_EDGES + 7) / 8;        // 400000
    const int rblocks = (N_NODES + 7) / 8;        // 12500

    gemm1_wmma<<<mblocks, 256, 0, stream>>>(x, W1, b1, H1);
    spmm128_atomic<<<eblocks, 256, 0, stream>>>(erow, ecol, eval, H1, S1);
    gemm2_wmma<<<mblocks, 256, 0, stream>>>(S1, W2, b2, O2);
    spmm40_atomic<<<eblocks, 256, 0, stream>>>(erow, ecol, eval, O2, S2);
    logsoftmax40<<<rblocks, 256, 0, stream>>>(S2, out);
}